// MF_17179869554
// MI455X (gfx1250) — compile-verified
//
#include <hip/hip_runtime.h>

typedef __attribute__((ext_vector_type(2))) float v2f;
typedef __attribute__((ext_vector_type(8))) float v8f;

// Reference: user_idx = x[r][0] + 0 ; item_idx = x[r][1] + N_USERS
#define OFFS_ITEM 1000000LL
#define EMBED_DIM 64

// One wave (32 lanes) computes 16 dot products via 16 accumulating
// V_WMMA_F32_16X16X4_F32 ops.
//
// A (16x64 user tile) per-WMMA-chunk j (K = 4j..4j+3), 32-bit A 16x4 layout:
//   lanes 0-15 : VGPR0 = A[m=lane][4j+0], VGPR1 = A[m=lane][4j+1]
//   lanes16-31 : VGPR0 = A[m=lane-16][4j+2], VGPR1 = A[m=lane-16][4j+3]
// B (64x16 item tile, B[k][n] = item[n][k]) has the *same* per-lane layout
// (lane = N = item row), so both operands are one coalesced b64 load per lane:
//   addr = row_ptr[lane%16] + j*16 + (lane>>4)*8 bytes.
__global__ __launch_bounds__(256) void mf_dot_wmma(
    const long long* __restrict__ x,      // (BATCH, 2) int64
    const float* __restrict__ table,      // (2e6, 64) fp32
    float* __restrict__ out)              // (BATCH,) fp32
{
    const int lane        = threadIdx.x & 31;
    const int waveInBlock = threadIdx.x >> 5;
    const int tile        = blockIdx.x * 8 + waveInBlock; // 16 rows per tile
    const int base        = tile * 16;

    const int rowInTile = lane & 15;
    const int half      = lane >> 4;     // 0: K%4 in {0,1}, 1: K%4 in {2,3}

    // Per-lane row indices (both lane halves service the same row).
    const long long* xp  = x + 2 * (long long)(base + rowInTile);
    const long long uidx = xp[0];               // + 0
    const long long iidx = xp[1] + OFFS_ITEM;   // + N_USERS

    const float* __restrict__ up = table + (size_t)uidx * EMBED_DIM + half * 2;
    const float* __restrict__ ip = table + (size_t)iidx * EMBED_DIM + half * 2;

    v8f acc = {};
#pragma unroll
    for (int j = 0; j < 16; ++j) {
        v2f a = *(const v2f*)(up + j * 4);   // global_load_b64, fully coalesced
        v2f b = *(const v2f*)(ip + j * 4);
        // 8 args: (neg_a, A, neg_b, B, c_mod, C, reuse_a, reuse_b)
        acc = __builtin_amdgcn_wmma_f32_16x16x4_f32(
                  false, a, false, b, (short)0, acc, false, false);
    }

    // D layout: VGPR v -> M=v (lanes 0-15, N=lane), M=v+8 (lanes 16-31, N=lane-16).
    // diag(m<8)  = acc[m]   @ lane m
    // diag(m>=8) = acc[m-8] @ lane m+16  (N = m -> lane = m+16, i.e. lane v+24)
    float dval = 0.0f;
#pragma unroll
    for (int v = 0; v < 8; ++v) {
        dval = (lane == v)      ? acc[v] : dval;
        dval = (lane == v + 24) ? acc[v] : dval;
    }
    if (lane < 8)
        out[base + lane] = dval;            // rows base+0 .. base+7
    else if (lane >= 24)
        out[base + lane - 16] = dval;       // rows base+8 .. base+15
}

extern "C" void kernel_launch(void* const* d_in, const int* in_sizes, int n_in,
                              void* d_out, int out_size, void* d_ws, size_t ws_size,
                              hipStream_t stream) {
    const long long* x     = (const long long*)d_in[0]; // (BATCH,2) int64
    const float*     table = (const float*)d_in[1];     // (2e6,64) fp32
    float*           out   = (float*)d_out;             // BATCH fp32

    const int batch  = out_size;        // 262144
    const int tiles  = batch / 16;      // 16384 waves
    const int blocks = tiles / 8;       // 8 waves (256 thr) per block -> 2048
    mf_dot_wmma<<<blocks, 256, 0, stream>>>(x, table, out);
}